// _GroundingDinoMultiscaleDeformableAttention_72808285602312
// MI455X (gfx1250) — compile-verified
//
#include <hip/hip_runtime.h>
#include <math.h>

// Problem constants (fixed by the reference module)
#define B_  8
#define Q_  900
#define D_  256
#define NH_ 8
#define NL_ 4
#define NP_ 4
#define S_  21760
#define HD_ 32

typedef float v2f __attribute__((ext_vector_type(2)));
typedef float v8f __attribute__((ext_vector_type(8)));

// ---------------------------------------------------------------------------
// hs = hidden + pos (float4 vectorized; n % 4 == 0)
// ---------------------------------------------------------------------------
__global__ void add_pos_kernel(const float4* __restrict__ a,
                               const float4* __restrict__ b,
                               float4* __restrict__ o, int n4) {
    int i = blockIdx.x * blockDim.x + threadIdx.x;
    if (i < n4) {
        float4 x = a[i], y = b[i];
        o[i] = make_float4(x.x + y.x, x.y + y.y, x.z + y.z, x.w + y.w);
    }
}

// ---------------------------------------------------------------------------
// LDS double-buffered WMMA-f32 GEMM: Dst[M,N] = A[M,K] @ W[K,N] + bias[N]
// Block: 128 threads = 4 waves; tile BM=128 x BN=64, K chunk BK=32.
// Wave w computes rows [w*32, w*32+32): TWO 16-row WMMA tiles x 64 cols
// (8 v8f accumulators), so each B fragment is reused by two WMMAs.
// Fragments (V_WMMA_F32_16X16X4_F32, wave32):
//   a[v]   = A[m][k + 2*half + v]      (ds_load_b64 from As)
//   b[v]   = W[k + 2*half + v][n]      (ds_load_2addr_b32 from Bs)
//   acc[r] = D[r + 8*half][n]
// Requires: N % 64 == 0, K % 32 == 0. M arbitrary (clamped staging, guarded
// stores; guards come after all WMMAs so EXEC is all-1s at every v_wmma).
// ---------------------------------------------------------------------------
#define BM 128
#define BN 64
#define BK 32
#define A_STRIDE 34   // even -> 8B-aligned b64 a-fragment loads
#define B_STRIDE 72   // 2*72 % 64 == 16 -> lane halves hit disjoint bank sets

__global__ __launch_bounds__(128) void gemm_bias_wmma(
        const float* __restrict__ A, const float* __restrict__ W,
        const float* __restrict__ bias, float* __restrict__ Dst,
        int M, int N, int K) {
    __shared__ float As[2][BM * A_STRIDE];   // 34816 B
    __shared__ float Bs[2][BK * B_STRIDE];   // 18432 B

    const int tid    = threadIdx.x;
    const int waveId = tid >> 5;             // 0..3
    const int lane   = tid & 31;

    const int blocksN = N / BN;
    const int bm = blockIdx.x / blocksN;
    const int bn = blockIdx.x - bm * blocksN;
    const int mBase = bm * BM;
    const int nBase = bn * BN;

    // staging decomposition (128 threads)
    const int aRow  = tid >> 3;         // 0..15, 8 passes of 16 rows
    const int aQuad = (tid & 7) * 4;    // k offset within chunk
    const int bRow  = tid >> 4;         // 0..7,  4 passes of 8 k-rows
    const int bQuad = (tid & 15) * 4;   // col offset within BN

    const int ml   = lane & 15;
    const int half = lane >> 4;

    float4 aReg[8];
    float4 bReg[4];

    const int nChunks = K / BK;

    auto loadChunk = [&](int kBase) {
#pragma unroll
        for (int p = 0; p < 8; ++p) {
            int row = mBase + aRow + p * 16;
            if (row >= M) row = M - 1;                    // clamp (harmless dup)
            aReg[p] = *(const float4*)(A + (size_t)row * K + kBase + aQuad);
        }
#pragma unroll
        for (int p = 0; p < 4; ++p) {
            const int krow = kBase + bRow + p * 8;
            bReg[p] = *(const float4*)(W + (size_t)krow * N + nBase + bQuad);
        }
    };
    auto storeChunk = [&](int buf) {
#pragma unroll
        for (int p = 0; p < 8; ++p) {
            float* d = &As[buf][(aRow + p * 16) * A_STRIDE + aQuad];
            d[0] = aReg[p].x; d[1] = aReg[p].y; d[2] = aReg[p].z; d[3] = aReg[p].w;
        }
#pragma unroll
        for (int p = 0; p < 4; ++p) {
            float* d = &Bs[buf][(bRow + p * 8) * B_STRIDE + bQuad];
            d[0] = bReg[p].x; d[1] = bReg[p].y; d[2] = bReg[p].z; d[3] = bReg[p].w;
        }
    };

    v8f acc[2][4];
#pragma unroll
    for (int s = 0; s < 2; ++s)
#pragma unroll
        for (int t = 0; t < 4; ++t) acc[s][t] = v8f{};

    loadChunk(0);
    storeChunk(0);
    __syncthreads();

    for (int c = 0; c < nChunks; ++c) {
        const int buf = c & 1;
        if (c + 1 < nChunks) loadChunk((c + 1) * BK);    // overlaps compute below

        const float* Abuf = &As[buf][(waveId * 32 + ml) * A_STRIDE + 2 * half];
        const float* Bbuf = &Bs[buf][2 * half * B_STRIDE + ml];
#pragma unroll
        for (int kk = 0; kk < BK; kk += 4) {
            v2f a0 = *(const v2f*)(Abuf + kk);                       // rows +0..15
            v2f a1 = *(const v2f*)(Abuf + 16 * A_STRIDE + kk);       // rows +16..31
            const float* bp = Bbuf + kk * B_STRIDE;
#pragma unroll
            for (int t = 0; t < 4; ++t) {
                v2f bf;
                bf.x = bp[t * 16];
                bf.y = bp[t * 16 + B_STRIDE];
                acc[0][t] = __builtin_amdgcn_wmma_f32_16x16x4_f32(
                    false, a0, false, bf, (short)0, acc[0][t], false, false);
                acc[1][t] = __builtin_amdgcn_wmma_f32_16x16x4_f32(
                    false, a1, false, bf, (short)0, acc[1][t], false, false);
            }
        }
        __syncthreads();
        if (c + 1 < nChunks) {
            storeChunk(buf ^ 1);
            __syncthreads();
        }
    }

    // epilogue: bias + guarded store (after all WMMAs)
#pragma unroll
    for (int s = 0; s < 2; ++s) {
        const int rowBase = mBase + waveId * 32 + s * 16;
#pragma unroll
        for (int t = 0; t < 4; ++t) {
            const int col = nBase + t * 16 + ml;
            const float bv = bias ? bias[col] : 0.0f;
#pragma unroll
            for (int r = 0; r < 8; ++r) {
                const int row = rowBase + r + 8 * half;
                if (row < M) Dst[(size_t)row * N + col] = acc[s][t][r] + bv;
            }
        }
    }
}

// ---------------------------------------------------------------------------
// In-place softmax over groups of 16 (the NL*NP axis). One thread per group.
// ---------------------------------------------------------------------------
__global__ void softmax16_kernel(float* __restrict__ p, int ngroups) {
    int i = blockIdx.x * blockDim.x + threadIdx.x;
    if (i >= ngroups) return;
    float* v = p + (size_t)i * 16;
    float mx = v[0];
#pragma unroll
    for (int j = 1; j < 16; ++j) mx = fmaxf(mx, v[j]);
    float e[16];
    float s = 0.0f;
#pragma unroll
    for (int j = 0; j < 16; ++j) { e[j] = expf(v[j] - mx); s += e[j]; }
    const float inv = 1.0f / s;
#pragma unroll
    for (int j = 0; j < 16; ++j) v[j] = e[j] * inv;
}

// ---------------------------------------------------------------------------
// Deformable sampling. One wave per (b,q,h); lane == channel (HD == 32).
// value:   (B, S, NH, HD)     attw: (B, Q, NH, NL*NP)  (post-softmax)
// offsets: (B, Q, NH, NL, NP, 2)   refp: (B, Q, NL, 2)
// out_pre: (B, Q, NH, HD) == (B, Q, D) row-major
// ---------------------------------------------------------------------------
__global__ __launch_bounds__(256) void ms_sample_kernel(
        const float* __restrict__ value, const float* __restrict__ attw,
        const float* __restrict__ offsets, const float* __restrict__ refp,
        float* __restrict__ out_pre) {
    const int wid = blockIdx.x * (blockDim.x >> 5) + (threadIdx.x >> 5);
    if (wid >= B_ * Q_ * NH_) return;        // wave-uniform
    const int lane = threadIdx.x & 31;

    const int h = wid % NH_;
    const int q = (wid / NH_) % Q_;
    const int b = wid / (NH_ * Q_);

    const int lvlStart[NL_] = {0, 128 * 128, 128 * 128 + 64 * 64,
                               128 * 128 + 64 * 64 + 32 * 32};
    const int lvlH[NL_] = {128, 64, 32, 16};
    const int lvlW[NL_] = {128, 64, 32, 16};

    float acc = 0.0f;
    const size_t bq = (size_t)b * Q_ + q;

    for (int l = 0; l < NL_; ++l) {
        const int   Hl = lvlH[l], Wl = lvlW[l];
        const float rx = refp[(bq * NL_ + l) * 2 + 0];
        const float ry = refp[(bq * NL_ + l) * 2 + 1];
        for (int p = 0; p < NP_; ++p) {
            const size_t oidx = ((((bq * NH_ + h) * NL_ + l) * NP_) + p) * 2;
            const float lx = rx + offsets[oidx + 0] / (float)Wl;
            const float ly = ry + offsets[oidx + 1] / (float)Hl;
            const float x = lx * (float)Wl - 0.5f;
            const float y = ly * (float)Hl - 0.5f;
            const float aw = attw[(bq * NH_ + h) * (NL_ * NP_) + l * NP_ + p];

            const float x0 = floorf(x), y0 = floorf(y);
            const float fx = x - x0,   fy = y - y0;
            const int ix0 = (int)x0,   iy0 = (int)y0;

            float s = 0.0f;
#pragma unroll
            for (int dy = 0; dy < 2; ++dy) {
#pragma unroll
                for (int dx = 0; dx < 2; ++dx) {
                    const int xi = ix0 + dx, yi = iy0 + dy;
                    const float wt = (dx ? fx : 1.0f - fx) * (dy ? fy : 1.0f - fy);
                    if (xi >= 0 && xi < Wl && yi >= 0 && yi < Hl) {
                        const size_t sidx = (size_t)lvlStart[l] + (size_t)yi * Wl + xi;
                        const size_t vidx =
                            (((size_t)b * S_ + sidx) * NH_ + h) * HD_ + lane;
                        s += value[vidx] * wt;
                    }
                }
            }
            acc += aw * s;
        }
    }
    out_pre[(bq * NH_ + h) * HD_ + lane] = acc;
}

// ---------------------------------------------------------------------------
// Host side
// ---------------------------------------------------------------------------
extern "C" void kernel_launch(void* const* d_in, const int* in_sizes, int n_in,
                              void* d_out, int out_size, void* d_ws, size_t ws_size,
                              hipStream_t stream) {
    const float* hidden = (const float*)d_in[0];
    const float* enc    = (const float*)d_in[1];
    const float* pos    = (const float*)d_in[2];
    const float* refp   = (const float*)d_in[3];
    /* d_in[4]: spatial_shapes (int32) — compile-time constants, unused */
    const float* Wv     = (const float*)d_in[5];
    const float* bv     = (const float*)d_in[6];
    const float* Woff   = (const float*)d_in[7];
    const float* boff   = (const float*)d_in[8];
    const float* Wattn  = (const float*)d_in[9];
    const float* battn  = (const float*)d_in[10];
    const float* Wout   = (const float*)d_in[11];
    const float* bout   = (const float*)d_in[12];

    float* out  = (float*)d_out;                        // (B, Q, D)
    float* attw = out + (size_t)B_ * Q_ * D_;           // (B, Q, NH, NL*NP)

    // Workspace layout
    char*  ws = (char*)d_ws;
    const size_t valueBytes = (size_t)B_ * S_ * D_ * sizeof(float);   // ~178 MB
    const size_t bqdBytes   = (size_t)B_ * Q_ * D_ * sizeof(float);   // ~7.4 MB
    float* value   = (float*)ws;                                      // (B,S,NH,HD)
    float* hs      = (float*)(ws + valueBytes);                       // (B,Q,D)
    float* offsets = (float*)(ws + valueBytes + bqdBytes);            // (B,Q,NH,NL,NP,2)
    float* out_pre = (float*)(ws + valueBytes + 2 * bqdBytes);        // (B,Q,NH,HD)

    auto launch_gemm = [&](const float* A, const float* W, const float* bias,
                           float* Dst, int M, int N, int K) {
        const int blocksM = (M + BM - 1) / BM;
        const int blocksN = N / BN;
        gemm_bias_wmma<<<blocksM * blocksN, 128, 0, stream>>>(A, W, bias, Dst,
                                                              M, N, K);
    };

    // 1. hs = hidden + pos
    {
        const int n4 = (B_ * Q_ * D_) / 4;
        add_pos_kernel<<<(n4 + 255) / 256, 256, 0, stream>>>(
            (const float4*)hidden, (const float4*)pos, (float4*)hs, n4);
    }
    // 2. value = enc @ W_value + b_value          (M=174080, N=256, K=256)
    launch_gemm(enc, Wv, bv, value, B_ * S_, D_, D_);
    // 3. offsets_raw = hs @ W_off + b_off         (M=7200, N=256, K=256)
    launch_gemm(hs, Woff, boff, offsets, B_ * Q_, NH_ * NL_ * NP_ * 2, D_);
    // 4. attn logits -> d_out attw region         (M=7200, N=128, K=256)
    launch_gemm(hs, Wattn, battn, attw, B_ * Q_, NH_ * NL_ * NP_, D_);
    // 5. softmax over NL*NP in place
    {
        const int ng = B_ * Q_ * NH_;
        softmax16_kernel<<<(ng + 255) / 256, 256, 0, stream>>>(attw, ng);
    }
    // 6. bilinear sampling + weighted sum -> out_pre
    {
        const int waves = B_ * Q_ * NH_;                 // one wave per (b,q,h)
        ms_sample_kernel<<<(waves + 7) / 8, 256, 0, stream>>>(value, attw, offsets,
                                                              refp, out_pre);
    }
    // 7. out = out_pre @ W_out + b_out            (M=7200, N=256, K=256)
    launch_gemm(out_pre, Wout, bout, out, B_ * Q_, D_, D_);
}